// GPM_31971736551497
// MI455X (gfx1250) — compile-verified
//
#include <hip/hip_runtime.h>
#include <cmath>

// ---------------------------------------------------------------------------
// GPM direct-writing forward for MI455X (gfx1250).
// Large GEMMs: 64x128 block tile, 32x32 per wave (4 WMMA accumulators),
// LDS double-buffered via GLOBAL_LOAD_ASYNC_TO_LDS_B128 (ASYNCcnt),
// compute on V_WMMA_F32_16X16X4_F32. Small GEMMs: direct-from-global WMMA.
// ---------------------------------------------------------------------------

typedef __attribute__((ext_vector_type(2))) float v2f;
typedef __attribute__((ext_vector_type(8))) float v8f;

__device__ __forceinline__ v8f wmma_f32(v2f a, v2f b, v8f c) {
    // 8 args: (neg_a, A, neg_b, B, c_mod, C, reuse_a, reuse_b)
    return __builtin_amdgcn_wmma_f32_16x16x4_f32(
        false, a, false, b, (short)0, c, false, false);
}

// Async copy 16 bytes global -> LDS (tracked by ASYNCcnt).
__device__ __forceinline__ void async_copy_b128(unsigned lds_byte, const float* g) {
    asm volatile("global_load_async_to_lds_b128 %0, %1, off"
                 :: "v"(lds_byte), "v"((unsigned long long)(uintptr_t)g)
                 : "memory");
}

__device__ __forceinline__ void wait_async_lds() {
    asm volatile("s_wait_asynccnt 0" ::: "memory");
}

// Stage A panel [64 x 32] and B panel [32 x 128] (f32) into LDS.
// A: 512 chunks of 16B (2/thread); B: 1024 chunks (4/thread).
__device__ __forceinline__ void stage_panels(
    const float* __restrict__ Ab, long long lda,
    const float* __restrict__ Bb, long long ldb,
    int k0, unsigned ldsA, unsigned ldsB, int tid)
{
#pragma unroll
    for (int i = 0; i < 2; ++i) {
        const int q = tid + i * 256;
        const int ar = q >> 3, ac = (q & 7) * 4;       // A: row, col4 within 32
        async_copy_b128(ldsA + (unsigned)q * 16u,
                        Ab + (long long)ar * lda + k0 + ac);
    }
#pragma unroll
    for (int i = 0; i < 4; ++i) {
        const int q = tid + i * 256;
        const int br = q >> 5, bc = (q & 31) * 4;      // B: k-row, col4 within 128
        async_copy_b128(ldsB + (unsigned)q * 16u,
                        Bb + (long long)(k0 + br) * ldb + bc);
    }
}

// One K=32 chunk of MMA from LDS. pa: [64][32], pb: [32][128].
// Wave computes 32x32: rows wmg*32..+31, cols wng*32..+31 of the block tile.
__device__ __forceinline__ void tile_mma(
    const float* pa, const float* pb,
    int wmg, int wng, int r, int hi,
    v8f& acc00, v8f& acc01, v8f& acc10, v8f& acc11)
{
    const float* la0 = pa + (wmg * 32 + r) * 32 + 2 * hi;   // rows +0..15
    const float* la1 = la0 + 16 * 32;                        // rows +16..31
    const float* lb = pb + wng * 32 + r + 2 * hi * 128;
#pragma unroll
    for (int kk = 0; kk < 32; kk += 4) {
        v2f a0, a1;
        a0.x = la0[kk];  a0.y = la0[kk + 1];
        a1.x = la1[kk];  a1.y = la1[kk + 1];
        const float* l = lb + kk * 128;
        v2f b0, b1;
        b0.x = l[0];   b0.y = l[128];
        b1.x = l[16];  b1.y = l[144];
        acc00 = wmma_f32(a0, b0, acc00);
        acc01 = wmma_f32(a0, b1, acc01);
        acc10 = wmma_f32(a1, b0, acc10);
        acc11 = wmma_f32(a1, b1, acc11);
    }
}

// LDS-staged batched GEMM: D = alpha*Cm + beta*(A@B).
// Requires M%64==0, N%128==0, K%32==0. One block per 64x128 tile.
// LDS float layout per buffer i (i=0,1), base offset i*6144:
//   A [0,2048) | B [2048,6144)
__global__ __launch_bounds__(256) void gemm_wmma_f32_lds(
    const float* __restrict__ A, long long lda, long long sA,
    const float* __restrict__ B, long long ldb, long long sB,
    const float* __restrict__ Cm, long long ldc, long long sC,
    float* __restrict__ D, long long ldd, long long sD,
    int M, int N, int K, int batch, float alpha, float beta)
{
    __shared__ float smem[12288];      // 48 KB: (A 2048 + B 4096) x 2 buffers

    const int tilesM = M >> 6;
    const int tilesN = N >> 7;
    const int per = tilesM * tilesN;
    const int bid = blockIdx.x;
    const int b = bid / per;
    const int t = bid - b * per;
    const int tm = t % tilesM;
    const int tn = t / tilesM;

    const int tid = threadIdx.x;
    const int lane = tid & 31;
    const int wave = tid >> 5;
    const int wmg = wave >> 2;         // 2 wave-rows x 32 = 64 M
    const int wng = wave & 3;          // 4 wave-cols x 32 = 128 N
    const int hi = (lane >> 4) & 1;
    const int r = lane & 15;

    const float* Ab = A + (long long)b * sA + (long long)(tm * 64) * lda;
    const float* Bb = B + (long long)b * sB + (long long)(tn * 128);

    const unsigned base = (unsigned)(uintptr_t)smem;   // low 32 bits = LDS offset

    v8f acc00 = {0.f, 0.f, 0.f, 0.f, 0.f, 0.f, 0.f, 0.f};
    v8f acc01 = {0.f, 0.f, 0.f, 0.f, 0.f, 0.f, 0.f, 0.f};
    v8f acc10 = {0.f, 0.f, 0.f, 0.f, 0.f, 0.f, 0.f, 0.f};
    v8f acc11 = {0.f, 0.f, 0.f, 0.f, 0.f, 0.f, 0.f, 0.f};

    const int nk = K >> 5;
    stage_panels(Ab, lda, Bb, ldb, 0, base, base + 8192u, tid);
    wait_async_lds();
    __syncthreads();

    int cur = 0;
    for (int c = 0; c < nk; ++c) {
        const int nxt = cur ^ 1;
        if (c + 1 < nk) {
            const unsigned oA = base + (unsigned)nxt * 24576u;
            const unsigned oB = oA + 8192u;
            stage_panels(Ab, lda, Bb, ldb, (c + 1) << 5, oA, oB, tid);
        }
        const float* pa = smem + (cur ? 6144 : 0);
        const float* pb = pa + 2048;
        tile_mma(pa, pb, wmg, wng, r, hi, acc00, acc01, acc10, acc11);
        wait_async_lds();              // our writes into buf[nxt] landed
        __syncthreads();               // all reads of buf[cur] + all writes done
        cur = nxt;
    }

    const int mBase = tm * 64 + wmg * 32 + hi * 8;
    const int n0 = tn * 128 + wng * 32 + r;
#pragma unroll
    for (int v = 0; v < 8; ++v) {
        const int m0 = mBase + v;
        const int m1 = m0 + 16;
        const long long i0 = (long long)b * sD + (long long)m0 * ldd + n0;
        const long long i1 = (long long)b * sD + (long long)m1 * ldd + n0;
        float c00 = 0.f, c01 = 0.f, c10 = 0.f, c11 = 0.f;
        if (Cm) {
            const long long j0 = (long long)b * sC + (long long)m0 * ldc + n0;
            const long long j1 = (long long)b * sC + (long long)m1 * ldc + n0;
            c00 = Cm[j0];
            c01 = Cm[j0 + 16];
            c10 = Cm[j1];
            c11 = Cm[j1 + 16];
        }
        D[i0]      = alpha * c00 + beta * acc00[v];
        D[i0 + 16] = alpha * c01 + beta * acc01[v];
        D[i1]      = alpha * c10 + beta * acc10[v];
        D[i1 + 16] = alpha * c11 + beta * acc11[v];
    }
}

// Direct-from-global fallback for small shapes (M%16==0, N%32==0, K%4==0).
__global__ __launch_bounds__(256) void gemm_wmma_f32(
    const float* __restrict__ A, long long lda, long long sA,
    const float* __restrict__ B, long long ldb, long long sB,
    const float* __restrict__ Cm, long long ldc, long long sC,
    float* __restrict__ D, long long ldd, long long sD,
    int M, int N, int K, int batch, float alpha, float beta)
{
    const int tilesM = M >> 4;
    const int tilesN = N >> 5;
    const int tilesPerBatch = tilesM * tilesN;
    const int wave = (int)((blockIdx.x * blockDim.x + threadIdx.x) >> 5);
    if (wave >= tilesPerBatch * batch) return;

    const int b  = wave / tilesPerBatch;
    const int t  = wave - b * tilesPerBatch;
    const int tm = t % tilesM;
    const int tn = t / tilesM;

    const int lane = threadIdx.x & 31;
    const int hi   = (lane >> 4) & 1;
    const int r    = lane & 15;

    const float* Ap = A + (long long)b * sA
                        + (long long)(tm * 16 + r) * lda + 2 * hi;
    const float* Bp = B + (long long)b * sB
                        + (long long)(tn * 32) + r + (long long)(2 * hi) * ldb;

    v8f acc0 = {0.f, 0.f, 0.f, 0.f, 0.f, 0.f, 0.f, 0.f};
    v8f acc1 = {0.f, 0.f, 0.f, 0.f, 0.f, 0.f, 0.f, 0.f};

    for (int k = 0; k < K; k += 4) {
        v2f a;
        a.x = Ap[k];
        a.y = Ap[k + 1];
        const float* Bk = Bp + (long long)k * ldb;
        v2f b0, b1;
        b0.x = Bk[0];
        b0.y = Bk[ldb];
        b1.x = Bk[16];
        b1.y = Bk[ldb + 16];
        acc0 = wmma_f32(a, b0, acc0);
        acc1 = wmma_f32(a, b1, acc1);
    }

    const int mBase = tm * 16 + hi * 8;
    const int n0 = tn * 32 + r;
#pragma unroll
    for (int v = 0; v < 8; ++v) {
        const int m = mBase + v;
        const long long i0 = (long long)b * sD + (long long)m * ldd + n0;
        float c0 = 0.f, c1 = 0.f;
        if (Cm) {
            const long long j0 = (long long)b * sC + (long long)m * ldc + n0;
            c0 = Cm[j0];
            c1 = Cm[j0 + 16];
        }
        D[i0]      = alpha * c0 + beta * acc0[v];
        D[i0 + 16] = alpha * c1 + beta * acc1[v];
    }
}

// out[b][j][i] = scale * in[b][i][j]   (in: rowsIn x colsIn)
__global__ __launch_bounds__(256) void transpose_scale_f32(
    const float* __restrict__ in, long long sIn, long long ldIn,
    float* __restrict__ out, long long sOut, long long ldOut,
    int rowsIn, int colsIn, int batch, float scale)
{
    const long long per = (long long)rowsIn * colsIn;
    const long long id = (long long)blockIdx.x * blockDim.x + threadIdx.x;
    if (id >= per * batch) return;
    const int b = (int)(id / per);
    const long long t = id - (long long)b * per;
    const int j = (int)(t / rowsIn);                 // column of in
    const int i = (int)(t - (long long)j * rowsIn);  // row of in
    out[(long long)b * sOut + (long long)j * ldOut + i] =
        scale * in[(long long)b * sIn + (long long)i * ldIn + j];
}

extern "C" void kernel_launch(void* const* d_in, const int* in_sizes, int n_in,
                              void* d_out, int out_size, void* d_ws, size_t ws_size,
                              hipStream_t stream) {
    (void)in_sizes; (void)n_in; (void)out_size; (void)ws_size;

    const float* input = (const float*)d_in[0];   // [E,B,C] = [32,64,768]
    const float* mem   = (const float*)d_in[1];   // [K,C]   = [512,768]
    float* out = (float*)d_out;                   // [E,B,C] f32
    float* ws  = (float*)d_ws;

    const int E = 32, Bn = 64, C = 768, K = 512;
    const long long BC = (long long)Bn * C;
    const float cinit = expf(-5.0f);

    size_t off = 0;
    auto alloc = [&](size_t n) { float* p = ws + off; off += n; return p; };
    float* Xa  = alloc((size_t)C * K);
    float* Xb  = alloc((size_t)C * K);
    float* T0  = alloc((size_t)K * K);
    float* W   = alloc((size_t)Bn * E * K);
    float* WPa = alloc((size_t)Bn * K * E);
    float* WPb = alloc((size_t)Bn * K * E);
    float* TW  = alloc((size_t)Bn * E * E);
    float* MN  = alloc((size_t)Bn * K * C);
    float* MPa = alloc((size_t)Bn * C * K);
    float* MPb = alloc((size_t)Bn * C * K);
    float* TM  = alloc((size_t)Bn * K * K);
    float* WR  = alloc((size_t)Bn * E * K);

    auto gemm = [&](const float* A, long long lda, long long sA,
                    const float* B, long long ldb, long long sB,
                    const float* Cmat, long long ldc, long long sC,
                    float* D, long long ldd, long long sD,
                    int M, int N, int Kd, int batch, float alpha, float beta) {
        if ((M % 64 == 0) && (N % 128 == 0) && (Kd % 32 == 0)) {
            const int blocks = (M / 64) * (N / 128) * batch;
            gemm_wmma_f32_lds<<<blocks, 256, 0, stream>>>(
                A, lda, sA, B, ldb, sB, Cmat, ldc, sC, D, ldd, sD,
                M, N, Kd, batch, alpha, beta);
        } else {
            const int tiles = (M / 16) * (N / 32) * batch;
            const int blocks = (tiles + 7) / 8;
            gemm_wmma_f32<<<blocks, 256, 0, stream>>>(
                A, lda, sA, B, ldb, sB, Cmat, ldc, sC, D, ldd, sD,
                M, N, Kd, batch, alpha, beta);
        }
    };
    auto tscale = [&](const float* in, long long sIn, long long ldIn,
                      float* o, long long sOut, long long ldOut,
                      int rowsIn, int colsIn, int batch, float scale) {
        const long long tot = (long long)rowsIn * colsIn * batch;
        const int blocks = (int)((tot + 255) / 256);
        transpose_scale_f32<<<blocks, 256, 0, stream>>>(
            in, sIn, ldIn, o, sOut, ldOut, rowsIn, colsIn, batch, scale);
    };

    // ---- Stage 1: P0 = pinv(M0)  (M0 identical across batch -> batch=1)
    tscale(mem, 0, C, Xa, 0, K, K, C, 1, cinit);
    float* Xc = Xa; float* Xn = Xb;
    for (int it = 0; it < 3; ++it) {
        gemm(mem, C, 0, Xc, K, 0, nullptr, 0, 0, T0, K, 0, K, K, C, 1, 0.f, 1.f);
        gemm(Xc, K, 0, T0, K, 0, Xc, K, 0, Xn, K, 0, C, K, K, 1, 2.f, -1.f);
        float* tmp = Xc; Xc = Xn; Xn = tmp;
    }
    const float* P0 = Xc;

    // ---- Stage 2: w = zb @ P0   (zb via strides from input [E,B,C])
    gemm(input, BC, C, P0, K, 0, nullptr, 0, 0,
         W, K, (long long)E * K, E, K, C, Bn, 0.f, 1.f);

    // ---- Stage 3: w_pinv = pinv(w), batched
    tscale(W, (long long)E * K, K, WPa, (long long)K * E, E, E, K, Bn, cinit);
    Xc = WPa; Xn = WPb;
    for (int it = 0; it < 3; ++it) {
        gemm(W, K, (long long)E * K, Xc, E, (long long)K * E, nullptr, 0, 0,
             TW, E, (long long)E * E, E, E, K, Bn, 0.f, 1.f);
        gemm(Xc, E, (long long)K * E, TW, E, (long long)E * E,
             Xc, E, (long long)K * E, Xn, E, (long long)K * E,
             K, E, E, Bn, 2.f, -1.f);
        float* tmp = Xc; Xc = Xn; Xn = tmp;
    }
    const float* WP = Xc;

    // ---- Stage 4: M_new = w_pinv @ zb : [512,768] per batch
    gemm(WP, E, (long long)K * E, input, BC, C, nullptr, 0, 0,
         MN, C, (long long)K * C, K, C, E, Bn, 0.f, 1.f);

    // ---- Stage 5: pinv(M_new), batched (dominant ~77 G MACs)
    tscale(MN, (long long)K * C, C, MPa, (long long)C * K, K, K, C, Bn, cinit);
    Xc = MPa; Xn = MPb;
    for (int it = 0; it < 3; ++it) {
        gemm(MN, C, (long long)K * C, Xc, K, (long long)C * K, nullptr, 0, 0,
             TM, K, (long long)K * K, K, K, C, Bn, 0.f, 1.f);
        gemm(Xc, K, (long long)C * K, TM, K, (long long)K * K,
             Xc, K, (long long)C * K, Xn, K, (long long)C * K,
             C, K, K, Bn, 2.f, -1.f);
        float* tmp = Xc; Xc = Xn; Xn = tmp;
    }
    const float* MP = Xc;

    // ---- Stage 6: w_read = zb @ M_new_pinv
    gemm(input, BC, C, MP, K, (long long)C * K, nullptr, 0, 0,
         WR, K, (long long)E * K, E, K, C, Bn, 0.f, 1.f);

    // ---- Stage 7: z_read = w_read @ M_new, stored directly as [E,B,C]
    gemm(WR, K, (long long)E * K, MN, C, (long long)K * C, nullptr, 0, 0,
         out, BC, C, E, C, K, Bn, 0.f, 1.f);
}